// LinearHierarchicalLoss_4475355922682
// MI455X (gfx1250) — compile-verified
//
#include <hip/hip_runtime.h>
#include <stdint.h>
#include <math.h>

#define C_TOTAL   2728            // 8+32+128+512+2048
#define NCHUNK    682             // C_TOTAL * 4 bytes / 16 bytes per b128
#define DMAX      5
#define BROWS     4096
#define BLK       256
#define DEPTH_PARAM 0.5f
#define LEN_PEN     1.5f

// ---------------------------------------------------------------------------
// Kernel 1: one block per row.  Async-stage y_pred/y_true rows into LDS with
// global_load_async_to_lds_b128 (CDNA5 async path, ASYNCcnt), dual argmax via
// wave32 shuffles, then the per-row hierarchical loss on thread 0.
// ---------------------------------------------------------------------------
__global__ __launch_bounds__(BLK) void row_loss_kernel(
    const float* __restrict__ y_pred,
    const float* __restrict__ y_true,
    const float* __restrict__ class_weights,
    const int*   __restrict__ path_ids,    // [C,5]
    const int*   __restrict__ path_len,    // [C]
    const int*   __restrict__ sib_start,   // [C,5]
    const int*   __restrict__ sib_size,    // [C,5]
    float*       __restrict__ ws)          // [BROWS] per-row loss
{
    __shared__ __align__(16) float sPred[C_TOTAL];
    __shared__ __align__(16) float sTrue[C_TOTAL];
    __shared__ float wVp[8]; __shared__ int wIp[8];
    __shared__ float wVt[8]; __shared__ int wIt[8];

    const int tid = threadIdx.x;
    const int row = blockIdx.x;

    // Uniform 64-bit row base pointers -> SGPR pair for GVS-mode async loads.
    const uint64_t gP = (uint64_t)(uintptr_t)(y_pred + (size_t)row * C_TOTAL);
    const uint64_t gT = (uint64_t)(uintptr_t)(y_true + (size_t)row * C_TOTAL);
    // On amdgcn the low 32 bits of a flat pointer to LDS are the LDS byte
    // offset (aperture hi bits select LDS), which is what VDST wants.
    const unsigned ldsP = (unsigned)(uintptr_t)(&sPred[0]);
    const unsigned ldsT = (unsigned)(uintptr_t)(&sTrue[0]);

    for (int c = tid; c < NCHUNK; c += BLK) {
        unsigned gb = (unsigned)c * 16u;   // byte offset within row (16B chunks)
        asm volatile("global_load_async_to_lds_b128 %0, %1, %2"
                     :: "v"(ldsP + gb), "v"(gb), "s"(gP) : "memory");
        asm volatile("global_load_async_to_lds_b128 %0, %1, %2"
                     :: "v"(ldsT + gb), "v"(gb), "s"(gT) : "memory");
    }
    asm volatile("s_wait_asynccnt 0" ::: "memory");   // my wave's copies done
    __syncthreads();                                  // all waves' copies visible

    // ---- dual argmax over LDS (first-occurrence semantics) ----
    float bpV = -INFINITY; int bpI = 0;
    float btV = -INFINITY; int btI = 0;
    for (int i = tid; i < C_TOTAL; i += BLK) {
        float v = sPred[i];
        if (v > bpV) { bpV = v; bpI = i; }   // per-thread i increases: strict >
        float w = sTrue[i];
        if (w > btV) { btV = w; btI = i; }
    }
    // wave32 reduction (value max, smaller index wins ties)
    #pragma unroll
    for (int off = 16; off > 0; off >>= 1) {
        float ov = __shfl_down(bpV, off, 32);
        int   oi = __shfl_down(bpI, off, 32);
        if (ov > bpV || (ov == bpV && oi < bpI)) { bpV = ov; bpI = oi; }
        float tv = __shfl_down(btV, off, 32);
        int   ti = __shfl_down(btI, off, 32);
        if (tv > btV || (tv == btV && ti < btI)) { btV = tv; btI = ti; }
    }
    const int lane = tid & 31, wave = tid >> 5;
    if (lane == 0) { wVp[wave] = bpV; wIp[wave] = bpI; wVt[wave] = btV; wIt[wave] = btI; }
    __syncthreads();

    if (tid == 0) {
        float pv = wVp[0]; int pi = wIp[0];
        float tv = wVt[0]; int ti = wIt[0];
        #pragma unroll
        for (int w = 1; w < 8; ++w) {
            if (wVp[w] > pv || (wVp[w] == pv && wIp[w] < pi)) { pv = wVp[w]; pi = wIp[w]; }
            if (wVt[w] > tv || (wVt[w] == tv && wIt[w] < ti)) { tv = wVt[w]; ti = wIt[w]; }
        }
        const int p = pi, t = ti;
        const int lp = path_len[p];
        const int lt = path_len[t];
        const int lmin = (lp < lt) ? lp : lt;

        float local = 0.0f;
        for (int l = 0; l < lmin; ++l) {
            const int st = sib_start[t * DMAX + l];
            const int sz = sib_size [t * DMAX + l];
            const int id = path_ids [t * DMAX + l];
            // masked-LSE: in-block logits + (C - sz) zero entries (exp(0)=1 each)
            float m = 0.0f;                        // max over {0, block values}
            for (int j = 0; j < sz; ++j) m = fmaxf(m, sPred[st + j]);
            float s = (float)(C_TOTAL - sz) * expf(-m);
            for (int j = 0; j < sz; ++j) s += expf(sPred[st + j] - m);
            const float lse = m + logf(s);
            const float ce  = lse - sPred[id];     // true id is inside the block
            const float h   = (float)(lt - l - 1);
            local += expf(-DEPTH_PARAM * h) * ce;
        }
        const float diff = fabsf((float)(lp - lt));
        ws[row] = local * (LEN_PEN * diff) * class_weights[t];
    }
}

// ---------------------------------------------------------------------------
// Kernel 2: deterministic fixed-order reduction of 4096 partials -> mean.
// ---------------------------------------------------------------------------
__global__ __launch_bounds__(1024) void reduce_mean_kernel(
    const float* __restrict__ ws, float* __restrict__ out)
{
    __shared__ float s[1024];
    const int tid = threadIdx.x;
    float v = ws[tid] + ws[tid + 1024] + ws[tid + 2048] + ws[tid + 3072];
    s[tid] = v;
    __syncthreads();
    #pragma unroll
    for (int off = 512; off > 0; off >>= 1) {
        if (tid < off) s[tid] += s[tid + off];
        __syncthreads();
    }
    if (tid == 0) out[0] = s[0] * (1.0f / (float)BROWS);
}

extern "C" void kernel_launch(void* const* d_in, const int* in_sizes, int n_in,
                              void* d_out, int out_size, void* d_ws, size_t ws_size,
                              hipStream_t stream) {
    const float* y_pred       = (const float*)d_in[0];
    const float* y_true       = (const float*)d_in[1];
    const float* class_weights= (const float*)d_in[2];
    const int*   path_ids     = (const int*)  d_in[3];
    const int*   path_len     = (const int*)  d_in[4];
    const int*   sib_start    = (const int*)  d_in[5];
    const int*   sib_size     = (const int*)  d_in[6];
    float* ws  = (float*)d_ws;
    float* out = (float*)d_out;

    row_loss_kernel<<<dim3(BROWS), dim3(BLK), 0, stream>>>(
        y_pred, y_true, class_weights, path_ids, path_len, sib_start, sib_size, ws);
    reduce_mean_kernel<<<dim3(1), dim3(1024), 0, stream>>>(ws, out);
}